// IntraInterCommunityBlock_61804579389716
// MI455X (gfx1250) — compile-verified
//
#include <hip/hip_runtime.h>
#include <math.h>

#define N_NODES 10000
#define C_DIM   512
#define E_EDGES 160000
#define KCHUNK  128
#define LDS_STRIDE 130   // 128 + 2 pad: avoids 16-way LDS bank conflicts on row reads

typedef float v2f __attribute__((ext_vector_type(2)));
typedef float v8f __attribute__((ext_vector_type(8)));

// ---------------------------------------------------------------------------
// Kernel 1: initialize accumulators with the residual (a = x) for both branches
// ---------------------------------------------------------------------------
__global__ __launch_bounds__(256) void init_acc_kernel(
    const float* __restrict__ x_re, const float* __restrict__ x_im,
    float* __restrict__ al_re, float* __restrict__ al_im,
    float* __restrict__ ag_re, float* __restrict__ ag_im)
{
    const size_t total = (size_t)N_NODES * C_DIM;
    size_t i = (size_t)blockIdx.x * blockDim.x + threadIdx.x;
    if (i < total) {
        float r = x_re[i];
        float m = x_im[i];
        al_re[i] = r; ag_re[i] = r;
        al_im[i] = m; ag_im[i] = m;
    }
}

// ---------------------------------------------------------------------------
// Kernel 2: fused complex GEMM for both branches using V_WMMA_F32_16X16X4_F32
//   hl_re = x_re@Wl_re^T - x_im@Wl_im^T + bl_re    (similarly hl_im, hg_*)
// Grid: (625 row tiles, 4 col groups). Block: 256 threads = 8 waves.
// Wave w computes the 16x16 tile at cols cg*128 + w*16, for all 4 outputs.
// x tiles staged in LDS (shared by the 8 waves); W read from global (L2-hot).
// ---------------------------------------------------------------------------
__global__ __launch_bounds__(256) void gemm_kernel(
    const float* __restrict__ x_re, const float* __restrict__ x_im,
    const float* __restrict__ Wl_re, const float* __restrict__ Wl_im,
    const float* __restrict__ Wg_re, const float* __restrict__ Wg_im,
    const float* __restrict__ bl_re, const float* __restrict__ bl_im,
    const float* __restrict__ bg_re, const float* __restrict__ bg_im,
    float* __restrict__ hl_re, float* __restrict__ hl_im,
    float* __restrict__ hg_re, float* __restrict__ hg_im)
{
    __shared__ float s_re[16 * LDS_STRIDE];
    __shared__ float s_im[16 * LDS_STRIDE];

    const int r0   = blockIdx.x * 16;           // node-row tile base (625 tiles)
    const int cg   = blockIdx.y;                // column group (0..3)
    const int t    = threadIdx.x;
    const int wave = t >> 5;                    // wave32
    const int lane = t & 31;
    const int half = lane >> 4;                 // 0: lanes 0-15, 1: lanes 16-31
    const int m    = lane & 15;
    const int n0   = cg * 128 + wave * 16;      // this wave's 16-col tile base
    const int ncol = n0 + m;

    v8f acc_lre = {}; v8f acc_lim = {};
    v8f acc_gre = {}; v8f acc_gim = {};

    const float* wlre_p = Wl_re + (size_t)ncol * C_DIM;
    const float* wlim_p = Wl_im + (size_t)ncol * C_DIM;
    const float* wgre_p = Wg_re + (size_t)ncol * C_DIM;
    const float* wgim_p = Wg_im + (size_t)ncol * C_DIM;

    for (int kc = 0; kc < C_DIM; kc += KCHUNK) {
        __syncthreads();  // protect LDS reuse from previous chunk
        // Cooperative stage of x_re/x_im 16x128 tiles: 512 float4s per array,
        // 256 threads -> 2 float4 each. Coalesced 16B global loads.
        #pragma unroll
        for (int i = 0; i < 2; ++i) {
            int idx = (t + i * 256) * 4;        // 0..2044
            int r   = idx >> 7;                 // row 0..15
            int c   = idx & 127;                // col 0..124
            const float4 vre = *(const float4*)(x_re + (size_t)(r0 + r) * C_DIM + kc + c);
            const float4 vim = *(const float4*)(x_im + (size_t)(r0 + r) * C_DIM + kc + c);
            float* dre = &s_re[r * LDS_STRIDE + c];
            float* dim_ = &s_im[r * LDS_STRIDE + c];
            // padded stride breaks 16B alignment -> two 8B stores each
            ((float2*)dre)[0]  = make_float2(vre.x, vre.y);
            ((float2*)(dre + 2))[0] = make_float2(vre.z, vre.w);
            ((float2*)dim_)[0] = make_float2(vim.x, vim.y);
            ((float2*)(dim_ + 2))[0] = make_float2(vim.z, vim.w);
        }
        __syncthreads();

        #pragma unroll 4
        for (int kk = 0; kk < KCHUNK; kk += 4) {
            const int ka = kk + 2 * half;       // per-lane K pair base
            // A fragments (16x4 f32 layout: VGPR j, lane l -> K = kk+j+2*half, M = l&15)
            v2f are = *(const v2f*)&s_re[m * LDS_STRIDE + ka];
            v2f aim = *(const v2f*)&s_im[m * LDS_STRIDE + ka];
            v2f aimn = -aim;                    // fp32 WMMA has no A-neg modifier
            // B fragments (4x16: B[k,n] = W[n,k], same per-lane pattern)
            const int kb = kc + ka;
            v2f blre = *(const v2f*)(wlre_p + kb);
            v2f blim = *(const v2f*)(wlim_p + kb);
            v2f bgre = *(const v2f*)(wgre_p + kb);
            v2f bgim = *(const v2f*)(wgim_p + kb);

            acc_lre = __builtin_amdgcn_wmma_f32_16x16x4_f32(false, are,  false, blre, (short)0, acc_lre, false, false);
            acc_lre = __builtin_amdgcn_wmma_f32_16x16x4_f32(false, aimn, false, blim, (short)0, acc_lre, false, false);
            acc_lim = __builtin_amdgcn_wmma_f32_16x16x4_f32(false, are,  false, blim, (short)0, acc_lim, false, false);
            acc_lim = __builtin_amdgcn_wmma_f32_16x16x4_f32(false, aim,  false, blre, (short)0, acc_lim, false, false);
            acc_gre = __builtin_amdgcn_wmma_f32_16x16x4_f32(false, are,  false, bgre, (short)0, acc_gre, false, false);
            acc_gre = __builtin_amdgcn_wmma_f32_16x16x4_f32(false, aimn, false, bgim, (short)0, acc_gre, false, false);
            acc_gim = __builtin_amdgcn_wmma_f32_16x16x4_f32(false, are,  false, bgim, (short)0, acc_gim, false, false);
            acc_gim = __builtin_amdgcn_wmma_f32_16x16x4_f32(false, aim,  false, bgre, (short)0, acc_gim, false, false);
        }
    }

    // Epilogue: add bias, store h tiles.
    // C/D layout: VGPR v, lane l -> M = v + 8*half, N = l&15.
    const float sblre = bl_re[ncol];
    const float sblim = bl_im[ncol];
    const float sbgre = bg_re[ncol];
    const float sbgim = bg_im[ncol];
    #pragma unroll
    for (int v = 0; v < 8; ++v) {
        const size_t o = (size_t)(r0 + v + 8 * half) * C_DIM + ncol;
        hl_re[o] = acc_lre[v] + sblre;
        hl_im[o] = acc_lim[v] + sblim;
        hg_re[o] = acc_gre[v] + sbgre;
        hg_im[o] = acc_gim[v] + sbgim;
    }
}

// ---------------------------------------------------------------------------
// Kernel 3: edge scatter. Each edge touches exactly ONE branch (intra if
// comm_id matches, else inter) — halves gather/scatter traffic vs reference.
// One block of 256 threads per edge; each thread handles 2 channels (float2).
// ---------------------------------------------------------------------------
__global__ __launch_bounds__(256) void scatter_kernel(
    const int* __restrict__ edge_index, const float* __restrict__ edge_weight,
    const int* __restrict__ comm_id,
    const float* __restrict__ hl_re, const float* __restrict__ hl_im,
    const float* __restrict__ hg_re, const float* __restrict__ hg_im,
    float* __restrict__ al_re, float* __restrict__ al_im,
    float* __restrict__ ag_re, float* __restrict__ ag_im)
{
    const int e = blockIdx.x;
    const int src = edge_index[e];
    const int dst = edge_index[E_EDGES + e];
    const float w = edge_weight[e];
    const bool same = (comm_id[src] == comm_id[dst]);

    const float* hre = same ? hl_re : hg_re;
    const float* him = same ? hl_im : hg_im;
    float* are = same ? al_re : ag_re;
    float* aim = same ? al_im : ag_im;

    const int c = threadIdx.x * 2;
    const size_t so = (size_t)src * C_DIM + c;
    const size_t dz = (size_t)dst * C_DIM + c;
    const float2 r = *(const float2*)(hre + so);
    const float2 i = *(const float2*)(him + so);
    atomicAdd(are + dz,     r.x * w);
    atomicAdd(are + dz + 1, r.y * w);
    atomicAdd(aim + dz,     i.x * w);
    atomicAdd(aim + dz + 1, i.y * w);
}

// ---------------------------------------------------------------------------
// Kernel 4: SiLU both branches, sum, interleave into (N, C, 2)
// ---------------------------------------------------------------------------
__device__ __forceinline__ float silu_f(float x) {
    return x / (1.0f + __expf(-x));
}

__global__ __launch_bounds__(256) void finalize_kernel(
    const float* __restrict__ al_re, const float* __restrict__ al_im,
    const float* __restrict__ ag_re, const float* __restrict__ ag_im,
    float* __restrict__ out)
{
    const size_t total = (size_t)N_NODES * C_DIM;
    size_t i = (size_t)blockIdx.x * blockDim.x + threadIdx.x;
    if (i < total) {
        float re = silu_f(al_re[i]) + silu_f(ag_re[i]);
        float im = silu_f(al_im[i]) + silu_f(ag_im[i]);
        out[2 * i]     = re;
        out[2 * i + 1] = im;
    }
}

// ---------------------------------------------------------------------------
extern "C" void kernel_launch(void* const* d_in, const int* in_sizes, int n_in,
                              void* d_out, int out_size, void* d_ws, size_t ws_size,
                              hipStream_t stream)
{
    const float* x_re  = (const float*)d_in[0];
    const float* x_im  = (const float*)d_in[1];
    const int*   eidx  = (const int*)  d_in[2];
    const float* ew    = (const float*)d_in[3];
    const int*   comm  = (const int*)  d_in[4];
    const float* Wl_re = (const float*)d_in[5];
    const float* Wl_im = (const float*)d_in[6];
    const float* bl_re = (const float*)d_in[7];
    const float* bl_im = (const float*)d_in[8];
    const float* Wg_re = (const float*)d_in[9];
    const float* Wg_im = (const float*)d_in[10];
    const float* bg_re = (const float*)d_in[11];
    const float* bg_im = (const float*)d_in[12];
    float* out = (float*)d_out;

    const size_t NC = (size_t)N_NODES * C_DIM;
    float* ws    = (float*)d_ws;
    float* hl_re = ws + 0 * NC;
    float* hl_im = ws + 1 * NC;
    float* hg_re = ws + 2 * NC;
    float* hg_im = ws + 3 * NC;
    float* al_re = ws + 4 * NC;
    float* al_im = ws + 5 * NC;
    float* ag_re = ws + 6 * NC;
    float* ag_im = ws + 7 * NC;

    const int elem_blocks = (int)((NC + 255) / 256);

    init_acc_kernel<<<elem_blocks, 256, 0, stream>>>(
        x_re, x_im, al_re, al_im, ag_re, ag_im);

    dim3 ggrid(N_NODES / 16, C_DIM / 128);  // 625 x 4
    gemm_kernel<<<ggrid, 256, 0, stream>>>(
        x_re, x_im, Wl_re, Wl_im, Wg_re, Wg_im,
        bl_re, bl_im, bg_re, bg_im,
        hl_re, hl_im, hg_re, hg_im);

    scatter_kernel<<<E_EDGES, 256, 0, stream>>>(
        eidx, ew, comm, hl_re, hl_im, hg_re, hg_im,
        al_re, al_im, ag_re, ag_im);

    finalize_kernel<<<elem_blocks, 256, 0, stream>>>(
        al_re, al_im, ag_re, ag_im, out);
}